// SAGE_68281390072709
// MI455X (gfx1250) — compile-verified
//
#include <hip/hip_runtime.h>
#include <hip/hip_bf16.h>

#define IN_FEATS 128
#define N_HIDDEN 128
#define N_CLASSES 41

typedef __attribute__((ext_vector_type(2))) float v2f;
typedef __attribute__((ext_vector_type(8))) float v8f;

// ---------------------------------------------------------------------------
// Utility: grid-stride zero fill
// ---------------------------------------------------------------------------
__global__ void sage_zero_kernel(float* __restrict__ p, long n) {
    long i = (long)blockIdx.x * blockDim.x + threadIdx.x;
    long stride = (long)gridDim.x * blockDim.x;
    for (; i < n; i += stride) p[i] = 0.0f;
}

// ---------------------------------------------------------------------------
// Degree: deg[dst[e]] += 1.0  (global_atomic_add_f32, no-return -> STOREcnt)
// ---------------------------------------------------------------------------
__global__ void sage_degree_kernel(const int* __restrict__ dst, float* __restrict__ deg,
                                   int n_edges) {
    int e = blockIdx.x * blockDim.x + threadIdx.x;
    if (e < n_edges) atomicAdd(deg + dst[e], 1.0f);
}

__global__ void sage_invdeg_kernel(const float* __restrict__ deg, float* __restrict__ invdeg,
                                   int n_nodes) {
    int i = blockIdx.x * blockDim.x + threadIdx.x;
    if (i < n_nodes) invdeg[i] = 1.0f / fmaxf(deg[i], 1.0f);
}

// ---------------------------------------------------------------------------
// Edge scatter-add: agg[dst] += feat[src].  One wave32 per edge: lane handles
// 4 consecutive floats -> the wave covers the full 128-float (512B) row with
// coalesced float4 loads; adds land as f32 atomics in L2 (h fits in 192MB L2).
// ---------------------------------------------------------------------------
__global__ void sage_scatter_kernel(const float* __restrict__ feat,
                                    const int* __restrict__ src,
                                    const int* __restrict__ dst,
                                    float* __restrict__ agg, int n_edges) {
    long tid = (long)blockIdx.x * blockDim.x + threadIdx.x;
    int  e   = (int)(tid >> 5);
    int  f   = ((int)tid & 31) << 2;
    if (e >= n_edges) return;
    int s = src[e];
    int d = dst[e];
    const float4 v = *reinterpret_cast<const float4*>(feat + (size_t)s * IN_FEATS + f);
    float* out = agg + (size_t)d * IN_FEATS + f;
    atomicAdd(out + 0, v.x);
    atomicAdd(out + 1, v.y);
    atomicAdd(out + 2, v.z);
    atomicAdd(out + 3, v.w);
}

// ---------------------------------------------------------------------------
// Layer 0: h = relu(x @ Wself + (agg*invdeg) @ Wneigh + b), all dims 128.
// Block = 256 threads = 8 waves; block owns 16 rows, wave w owns cols
// [16w, 16w+16).  Per wave: 64x v_wmma_f32_16x16x4_f32 in one f32 acc chain.
//
// CDNA5 f32 WMMA lane layout (ISA 7.12.2):
//   A 16x4: row = lane%16; VGPR0 = K (lanes<16) / K+2 (lanes>=16); VGPR1 = +1
//   B 4x16: col = lane%16; VGPR0 = row K / K+2;  VGPR1 = +1
//   C/D 16x16: col = lane%16; VGPR r = row r (lanes<16) / r+8 (lanes>=16)
// ---------------------------------------------------------------------------
__global__ __launch_bounds__(256) void sage_gemm_relu0_kernel(
    const float* __restrict__ x, const float* __restrict__ agg,
    const float* __restrict__ invdeg,
    const float* __restrict__ Wself, const float* __restrict__ Wneigh,
    const float* __restrict__ bias, float* __restrict__ h, int n_nodes) {
    const int lane  = threadIdx.x & 31;
    const int wave  = threadIdx.x >> 5;        // 0..7
    const int m0    = blockIdx.x * 16;
    const int n0    = wave * 16;
    const int l16   = lane & 15;
    const int khalf = (lane >> 4) << 1;        // 0 or 2
    const int col   = n0 + l16;

    int row = m0 + l16;
    row = row < n_nodes ? row : n_nodes - 1;   // branchless clamp: EXEC stays all-1

    const float idg = invdeg[row];
    const float* xr = x   + (size_t)row * IN_FEATS;
    const float* ar = agg + (size_t)row * IN_FEATS;

    v8f c = {};
#pragma unroll
    for (int k = 0; k < IN_FEATS; k += 4) {
        v2f a, b;
        a.x = xr[k + khalf];
        a.y = xr[k + khalf + 1];
        b.x = Wself[(size_t)(k + khalf)     * N_HIDDEN + col];
        b.y = Wself[(size_t)(k + khalf + 1) * N_HIDDEN + col];
        c = __builtin_amdgcn_wmma_f32_16x16x4_f32(false, a, false, b, (short)0, c,
                                                  false, false);
    }
#pragma unroll
    for (int k = 0; k < IN_FEATS; k += 4) {
        v2f a, b;
        a.x = ar[k + khalf] * idg;
        a.y = ar[k + khalf + 1] * idg;
        b.x = Wneigh[(size_t)(k + khalf)     * N_HIDDEN + col];
        b.y = Wneigh[(size_t)(k + khalf + 1) * N_HIDDEN + col];
        c = __builtin_amdgcn_wmma_f32_16x16x4_f32(false, a, false, b, (short)0, c,
                                                  false, false);
    }

    const float bb = bias[col];
#pragma unroll
    for (int r = 0; r < 8; ++r) {
        int orow = m0 + r + ((lane >> 4) << 3);
        float v = c[r] + bb;
        v = v > 0.0f ? v : 0.0f;
        if (orow < n_nodes) h[(size_t)orow * N_HIDDEN + col] = v;
    }
}

// ---------------------------------------------------------------------------
// Layer 1: out = h @ Wself1 + (agg*invdeg) @ Wneigh1 + b1  (K=128, N=41).
// 3 waves per block cover cols [0,48); cols >= 41 masked with branchless
// selects (clamped address + 0-multiply) so EXEC stays all-1 for WMMA.
// ---------------------------------------------------------------------------
__global__ __launch_bounds__(96) void sage_gemm1_kernel(
    const float* __restrict__ h, const float* __restrict__ agg,
    const float* __restrict__ invdeg,
    const float* __restrict__ Wself, const float* __restrict__ Wneigh,
    const float* __restrict__ bias, float* __restrict__ out, int n_nodes) {
    const int lane  = threadIdx.x & 31;
    const int wave  = threadIdx.x >> 5;        // 0..2
    const int m0    = blockIdx.x * 16;
    const int n0    = wave * 16;
    const int l16   = lane & 15;
    const int khalf = (lane >> 4) << 1;
    const int col   = n0 + l16;
    const bool cok  = col < N_CLASSES;
    const int  colc = cok ? col : (N_CLASSES - 1);
    const float cmask = cok ? 1.0f : 0.0f;

    int row = m0 + l16;
    row = row < n_nodes ? row : n_nodes - 1;

    const float idg = invdeg[row];
    const float* hr = h   + (size_t)row * N_HIDDEN;
    const float* ar = agg + (size_t)row * N_HIDDEN;

    v8f c = {};
#pragma unroll
    for (int k = 0; k < N_HIDDEN; k += 4) {
        v2f a, b;
        a.x = hr[k + khalf];
        a.y = hr[k + khalf + 1];
        b.x = Wself[(size_t)(k + khalf)     * N_CLASSES + colc] * cmask;
        b.y = Wself[(size_t)(k + khalf + 1) * N_CLASSES + colc] * cmask;
        c = __builtin_amdgcn_wmma_f32_16x16x4_f32(false, a, false, b, (short)0, c,
                                                  false, false);
    }
#pragma unroll
    for (int k = 0; k < N_HIDDEN; k += 4) {
        v2f a, b;
        a.x = ar[k + khalf] * idg;
        a.y = ar[k + khalf + 1] * idg;
        b.x = Wneigh[(size_t)(k + khalf)     * N_CLASSES + colc] * cmask;
        b.y = Wneigh[(size_t)(k + khalf + 1) * N_CLASSES + colc] * cmask;
        c = __builtin_amdgcn_wmma_f32_16x16x4_f32(false, a, false, b, (short)0, c,
                                                  false, false);
    }

    const float bb = bias[colc];
#pragma unroll
    for (int r = 0; r < 8; ++r) {
        int orow = m0 + r + ((lane >> 4) << 3);
        float v = c[r] + bb;
        if (cok && orow < n_nodes) out[(size_t)orow * N_CLASSES + col] = v;
    }
}

// ---------------------------------------------------------------------------
// Host launch
// ---------------------------------------------------------------------------
extern "C" void kernel_launch(void* const* d_in, const int* in_sizes, int n_in,
                              void* d_out, int out_size, void* d_ws, size_t ws_size,
                              hipStream_t stream) {
    const float* x    = (const float*)d_in[0];
    const int*   src  = (const int*)d_in[1];
    const int*   dst  = (const int*)d_in[2];
    const float* Ws0  = (const float*)d_in[3];
    const float* Wn0  = (const float*)d_in[4];
    const float* b0   = (const float*)d_in[5];
    const float* Ws1  = (const float*)d_in[6];
    const float* Wn1  = (const float*)d_in[7];
    const float* b1   = (const float*)d_in[8];
    float*       out  = (float*)d_out;

    const int n_nodes = in_sizes[0] / IN_FEATS;
    const int n_edges = in_sizes[1];

    // Workspace layout: agg | deg | invdeg | h   (~51.6 MB for this problem)
    float* agg    = (float*)d_ws;
    float* deg    = agg + (size_t)n_nodes * IN_FEATS;
    float* invdeg = deg + n_nodes;
    float* hbuf   = invdeg + n_nodes;

    const long aggN   = (long)n_nodes * IN_FEATS;
    const int  mtiles = (n_nodes + 15) / 16;
    const long scatThreads = (long)n_edges * 32;
    const int  scatBlocks  = (int)((scatThreads + 255) / 256);

    // ---- Layer 0 ----
    // zero agg + deg (adjacent)
    sage_zero_kernel<<<4096, 256, 0, stream>>>(agg, aggN + n_nodes);
    sage_degree_kernel<<<(n_edges + 255) / 256, 256, 0, stream>>>(dst, deg, n_edges);
    sage_invdeg_kernel<<<(n_nodes + 255) / 256, 256, 0, stream>>>(deg, invdeg, n_nodes);
    sage_scatter_kernel<<<scatBlocks, 256, 0, stream>>>(x, src, dst, agg, n_edges);
    sage_gemm_relu0_kernel<<<mtiles, 256, 0, stream>>>(x, agg, invdeg, Ws0, Wn0, b0,
                                                       hbuf, n_nodes);

    // ---- Layer 1 (reuse agg buffer) ----
    sage_zero_kernel<<<4096, 256, 0, stream>>>(agg, aggN);
    sage_scatter_kernel<<<scatBlocks, 256, 0, stream>>>(hbuf, src, dst, agg, n_edges);
    sage_gemm1_kernel<<<mtiles, 96, 0, stream>>>(hbuf, agg, invdeg, Ws1, Wn1, b1,
                                                 out, n_nodes);
}